// GIN_27384711480020
// MI455X (gfx1250) — compile-verified
//
#include <hip/hip_runtime.h>

#define N_NODES 50000
#define D 128
#define TM 16   // nodes per MLP block

typedef float v2f __attribute__((ext_vector_type(2)));
typedef float v8f __attribute__((ext_vector_type(8)));

// ---------------------------------------------------------------------------
// acc = src  (vectorized float4 copy; initializes the "+ x" term of GIN)
// ---------------------------------------------------------------------------
__global__ __launch_bounds__(256) void gin_copy_kernel(const float* __restrict__ src,
                                                       float* __restrict__ dst, int n4) {
    int i = blockIdx.x * blockDim.x + threadIdx.x;
    if (i < n4) {
        reinterpret_cast<float4*>(dst)[i] = reinterpret_cast<const float4*>(src)[i];
    }
}

// ---------------------------------------------------------------------------
// acc[dst] += x[src] over all edges. One wave32 per edge, each lane handles a
// float4 feature chunk (128 floats / 32 lanes). All atomics hit L2 (25.6 MB
// accumulator is L2-resident on the 192 MB L2).
// ---------------------------------------------------------------------------
__global__ __launch_bounds__(256) void gin_scatter_kernel(const float* __restrict__ x,
                                                          const int* __restrict__ src_idx,
                                                          const int* __restrict__ dst_idx,
                                                          float* __restrict__ acc, int nE) {
    int tid  = blockIdx.x * blockDim.x + threadIdx.x;
    int e    = tid >> 5;
    int lane = tid & 31;
    if (e >= nE) return;
    int s = src_idx[e];
    int d = dst_idx[e];
    const float4 v = *reinterpret_cast<const float4*>(x + (size_t)s * D + lane * 4);
    float* p = acc + (size_t)d * D + lane * 4;
    unsafeAtomicAdd(p + 0, v.x);
    unsafeAtomicAdd(p + 1, v.y);
    unsafeAtomicAdd(p + 2, v.z);
    unsafeAtomicAdd(p + 3, v.w);
}

// ---------------------------------------------------------------------------
// Fused GIN MLP:  out = [relu]( relu(X @ Wa + ba) @ Wb + bb )
// Block = 256 threads (8 waves) handles TM=16 nodes. Wave w computes the
// 16x16 output tile for columns [16w, 16w+16) using V_WMMA_F32_16X16X4_F32
// chained over K=128 (32 wmma per GEMM per wave).
//
// A-fragment (16x4 f32, ISA 7.12.2):  lane m16 = lane&15 holds row M=m16;
//   vgpr0 = K=k0+2*half, vgpr1 = K=k0+1+2*half  (half = lane>>4)
// B-fragment (4x16 f32, mirrored):    lane m16 holds col N;
//   vgpr0 = K=k0+2*half, vgpr1 = K=k0+1+2*half
// C/D (16x16 f32): lane m16 = N, element v holds M = v + 8*half
// ---------------------------------------------------------------------------
template <int RELU_OUT>
__global__ __launch_bounds__(256) void gin_mlp_kernel(const float* __restrict__ xin,
                                                      const float* __restrict__ Wa,
                                                      const float* __restrict__ ba,
                                                      const float* __restrict__ Wb,
                                                      const float* __restrict__ bb,
                                                      float* __restrict__ out) {
    __shared__ float XT[D][TM + 1];   // input tile, transposed, pad 17 -> conflict-free
    __shared__ float HT[D][TM + 1];   // hidden tile, transposed

    const int node_base = blockIdx.x * TM;
    const int t = threadIdx.x;

    // Stage 0: load 16x128 input tile transposed into LDS (coalesced global reads)
#pragma unroll
    for (int i = 0; i < (TM * D) / 256; ++i) {
        int e   = t + 256 * i;
        int row = e >> 7;        // / 128
        int col = e & (D - 1);
        XT[col][row] = xin[(size_t)(node_base + row) * D + col];
    }
    __syncthreads();

    const int w    = t >> 5;         // wave id -> N tile
    const int lane = t & 31;
    const int half = lane >> 4;
    const int m16  = lane & 15;
    const int ncol = w * 16 + m16;   // output column this lane owns

    // ---- Stage 1: H = relu(X @ Wa + ba) ----
    v8f c = {};
#pragma unroll 8
    for (int k0 = 0; k0 < D; k0 += 4) {
        const int ka = k0 + 2 * half;
        v2f a, b;
        a.x = XT[ka][m16];
        a.y = XT[ka + 1][m16];
        b.x = Wa[(size_t)ka * D + ncol];
        b.y = Wa[(size_t)(ka + 1) * D + ncol];
        c = __builtin_amdgcn_wmma_f32_16x16x4_f32(false, a, false, b, (short)0, c,
                                                  false, false);
    }
    {
        const float bias = ba[ncol];
#pragma unroll
        for (int v = 0; v < 8; ++v) {
            float r = c[v] + bias;
            r = r > 0.f ? r : 0.f;
            HT[ncol][v + 8 * half] = r;
        }
    }
    __syncthreads();

    // ---- Stage 2: out = H @ Wb + bb  (optional outer relu) ----
    c = (v8f){};
#pragma unroll 8
    for (int k0 = 0; k0 < D; k0 += 4) {
        const int ka = k0 + 2 * half;
        v2f a, b;
        a.x = HT[ka][m16];
        a.y = HT[ka + 1][m16];
        b.x = Wb[(size_t)ka * D + ncol];
        b.y = Wb[(size_t)(ka + 1) * D + ncol];
        c = __builtin_amdgcn_wmma_f32_16x16x4_f32(false, a, false, b, (short)0, c,
                                                  false, false);
    }
    {
        const float bias = bb[ncol];
#pragma unroll
        for (int v = 0; v < 8; ++v) {
            float r = c[v] + bias;
            if (RELU_OUT) r = r > 0.f ? r : 0.f;
            out[(size_t)(node_base + v + 8 * half) * D + ncol] = r;
        }
    }
}

// ---------------------------------------------------------------------------
extern "C" void kernel_launch(void* const* d_in, const int* in_sizes, int n_in,
                              void* d_out, int out_size, void* d_ws, size_t ws_size,
                              hipStream_t stream) {
    const float* x  = (const float*)d_in[0];
    const int*  ei  = (const int*)d_in[1];
    const float* W1 = (const float*)d_in[2];
    const float* b1 = (const float*)d_in[3];
    const float* W2 = (const float*)d_in[4];
    const float* b2 = (const float*)d_in[5];
    const float* W3 = (const float*)d_in[6];
    const float* b3 = (const float*)d_in[7];
    const float* W4 = (const float*)d_in[8];
    const float* b4 = (const float*)d_in[9];
    float* out = (float*)d_out;

    const int E = in_sizes[1] / 2;
    const int* src = ei;       // edge_index[0]
    const int* dst = ei + E;   // edge_index[1]

    float* acc = (float*)d_ws;                       // 50000*128 f32 = 25.6 MB
    float* h   = acc + (size_t)N_NODES * D;          // 25.6 MB

    const int n4       = N_NODES * D / 4;
    const int cpBlocks = (n4 + 255) / 256;
    const int scBlocks = (E * 32 + 255) / 256;
    const int mlpBlocks = N_NODES / TM;              // 3125, exact

    // Layer 1: h = relu( MLP1( x + segsum(x[src] -> dst) ) )
    gin_copy_kernel<<<cpBlocks, 256, 0, stream>>>(x, acc, n4);
    gin_scatter_kernel<<<scBlocks, 256, 0, stream>>>(x, src, dst, acc, E);
    gin_mlp_kernel<1><<<mlpBlocks, 256, 0, stream>>>(acc, W1, b1, W2, b2, h);

    // Layer 2: out = MLP2( h + segsum(h[src] -> dst) )
    gin_copy_kernel<<<cpBlocks, 256, 0, stream>>>(h, acc, n4);
    gin_scatter_kernel<<<scBlocks, 256, 0, stream>>>(h, src, dst, acc, E);
    gin_mlp_kernel<0><<<mlpBlocks, 256, 0, stream>>>(acc, W3, b3, W4, b4, out);
}